// DiscriminatorHierarchical0_82480551952950
// MI455X (gfx1250) — compile-verified
//
#include <hip/hip_runtime.h>

// ---------------------------------------------------------------------------
// DiscriminatorHierarchical for MI455X (gfx1250, wave32, WMMA).
//
// Sizes:  NODE = [212455, 32768, 8192, 2048, 512, 128]
//         CH   = [1, 8, 16, 32, 64, 128],  BATCH = 64
// Per layer:  h = z @ W ; y[:, parents, :] += h ; z = leaky_relu(y + b, 0.2)
// Final:      rf = sigmoid(transpose(z,(0,2,1)).flat @ fc_w + fc_b)
//
// Layers 1..4 use V_WMMA_F32_16X16X4_F32: one block per input node, 4 waves
// per block, wave w computes the (16-batch x C_out) tile for batches
// 16w..16w+15, accumulating over K in steps of 4, then scatter-adds into the
// parent node with global f32 atomics (parents are random -> no locality).
// ---------------------------------------------------------------------------

typedef __attribute__((ext_vector_type(2))) float v2f;
typedef __attribute__((ext_vector_type(8))) float v8f;

#define BATCH 64
#define NEG_SLOPE 0.2f

#define N0 212455
#define N1 32768
#define N2 8192
#define N3 2048
#define N4 512
#define N5 128

// ---------------------------------------------------------------- zero fill
__global__ void zero_f32(float* __restrict__ p, long long n) {
    long long i = (long long)blockIdx.x * blockDim.x + threadIdx.x;
    long long stride = (long long)gridDim.x * blockDim.x;
    for (; i < n; i += stride) p[i] = 0.0f;
}

// ------------------------------------------------- layer 0: C_in=1 -> C_out=8
// h[b,i,c] = x[b,i] * W0[c]; scatter into y[b, parents0[i], c]
__global__ void layer0_scatter(const float* __restrict__ x,
                               const float* __restrict__ W0,
                               const int*  __restrict__ parents,
                               float* __restrict__ y,
                               long long total) {
    float w[8];
#pragma unroll
    for (int c = 0; c < 8; ++c) w[c] = W0[c];

    long long i = (long long)blockIdx.x * blockDim.x + threadIdx.x;
    long long stride = (long long)gridDim.x * blockDim.x;
    for (; i < total; i += stride) {
        int node = (int)(i % N0);
        int b    = (int)(i / N0);
        float v  = x[i];
        int p    = parents[node];
        float* dst = y + ((long long)b * N1 + (long long)p) * 8;
#pragma unroll
        for (int c = 0; c < 8; ++c) atomicAdd(dst + c, v * w[c]);
    }
}

// --------------------------------------- layers 1..4: WMMA f32 16x16x4 GEMM
// z: (64, NIN, CIN)  W: (CIN, COUT)  y: (64, NOUT, COUT)
// grid = NIN blocks, 128 threads (4 waves); wave w -> batches 16w..16w+15.
template <int CIN, int COUT>
__global__ void fgl_wmma(const float* __restrict__ z,
                         const float* __restrict__ W,
                         const int*  __restrict__ parents,
                         float* __restrict__ y,
                         int nin, int nout) {
    const int node = blockIdx.x;           // uniform per block
    const int wv   = threadIdx.x >> 5;     // batch slab 0..3 (uniform per wave)
    const int lane = threadIdx.x & 31;
    const int m    = lane & 15;            // row-in-tile / column-in-tile
    const int half = lane >> 4;            // 0: K0,K1  1: K2,K3

    // ---- A fragments: 16x4 tiles over K, all K steps preloaded.
    // Lane layout (ISA 7.12.2, 32-bit A 16x4): lanes 0-15 -> M=lane, K={0,1};
    // lanes 16-31 -> M=lane-16, K={2,3}.
    const float* zrow = z + ((long long)(wv * 16 + m) * nin + node) * CIN;
    v2f afrag[CIN / 4];
#pragma unroll
    for (int k = 0; k < CIN / 4; ++k) {
        afrag[k].x = zrow[k * 4 + 2 * half + 0];
        afrag[k].y = zrow[k * 4 + 2 * half + 1];
    }

    const int p = parents[node];
    float* ybase = y + (long long)p * COUT;

#pragma unroll
    for (int nt = 0; nt < COUT / 16; ++nt) {
        v8f acc = {};
#pragma unroll
        for (int k = 0; k < CIN / 4; ++k) {
            // B fragment: 4x16 tile, row K striped across lanes (N = lane&15),
            // lanes 0-15 hold K={0,1}, lanes 16-31 hold K={2,3}.
            v2f bfrag;
            bfrag.x = W[(k * 4 + 2 * half + 0) * COUT + nt * 16 + m];
            bfrag.y = W[(k * 4 + 2 * half + 1) * COUT + nt * 16 + m];
            acc = __builtin_amdgcn_wmma_f32_16x16x4_f32(
                /*neg_a=*/false, afrag[k], /*neg_b=*/false, bfrag,
                /*c_mod=*/(short)0, acc, /*reuse_a=*/false, /*reuse_b=*/false);
        }
        // ---- scatter D: lane holds (M = r + 8*half, N = lane&15) in vgpr r.
#pragma unroll
        for (int r = 0; r < 8; ++r) {
            int ob = wv * 16 + r + 8 * half;          // batch index
            atomicAdd(ybase + (long long)ob * nout * COUT + nt * 16 + m,
                      acc[r]);
        }
    }
}

// ------------------------------------------------ bias + leaky ReLU in place
template <int COUT>
__global__ void bias_lrelu(float* __restrict__ y,
                           const float* __restrict__ bias, long long n) {
    long long i = (long long)blockIdx.x * blockDim.x + threadIdx.x;
    long long stride = (long long)gridDim.x * blockDim.x;
    for (; i < n; i += stride) {
        float v = y[i] + bias[(int)(i & (COUT - 1))];
        y[i] = v > 0.0f ? v : NEG_SLOPE * v;
    }
}

// ------------------------------------------------------- final FC + sigmoid
// z: (64, 128 nodes, 128 ch); flat[b, c*128+n] = z[b, n, c]
__global__ void fc_head(const float* __restrict__ z,
                        const float* __restrict__ fc_w,
                        const float* __restrict__ fc_b,
                        float* __restrict__ out) {
    __shared__ float red[256];
    const int b = blockIdx.x;
    const int t = threadIdx.x;
    float s = 0.0f;
    for (int j = t; j < 16384; j += 256) {
        int c = j >> 7, n = j & 127;
        s += z[(long long)b * 16384 + n * 128 + c] * fc_w[j];
    }
    red[t] = s;
    __syncthreads();
#pragma unroll
    for (int off = 128; off > 0; off >>= 1) {
        if (t < off) red[t] += red[t + off];
        __syncthreads();
    }
    if (t == 0) {
        float v = red[0] + fc_b[0];
        out[b]         = 1.0f / (1.0f + expf(-v));   // rf
        out[64 + b]    = -1.0f;
        out[128 + b]   = -1.0f;
        out[192 + b]   = -1.0f;
    }
}

// ---------------------------------------------------------------------------
extern "C" void kernel_launch(void* const* d_in, const int* in_sizes, int n_in,
                              void* d_out, int out_size, void* d_ws, size_t ws_size,
                              hipStream_t stream) {
    // setup_inputs() dict order:
    // x, (W0,b0,parents0), (W1,b1,parents1), ..., (W4,b4,parents4), fc_w, fc_b
    const float* x        = (const float*)d_in[0];
    const float* Wl[5]    = { (const float*)d_in[1], (const float*)d_in[4],
                              (const float*)d_in[7], (const float*)d_in[10],
                              (const float*)d_in[13] };
    const float* bl[5]    = { (const float*)d_in[2], (const float*)d_in[5],
                              (const float*)d_in[8], (const float*)d_in[11],
                              (const float*)d_in[14] };
    const int*   par[5]   = { (const int*)d_in[3], (const int*)d_in[6],
                              (const int*)d_in[9], (const int*)d_in[12],
                              (const int*)d_in[15] };
    const float* fc_w     = (const float*)d_in[16];
    const float* fc_b     = (const float*)d_in[17];
    float*       out      = (float*)d_out;

    // Workspace ping-pong: region A holds z0/z2/z4 (max 16.78M f32 = 64 MB),
    // region B holds z1/z3 (max 8.39M f32 = 32 MB). Total ~96 MB of d_ws.
    const long long szA = (long long)BATCH * N1 * 8;    // 16,777,216
    float* bufA = (float*)d_ws;
    float* bufB = bufA + szA;

    const long long y0n = (long long)BATCH * N1 * 8;    // layer0 out
    const long long y1n = (long long)BATCH * N2 * 16;
    const long long y2n = (long long)BATCH * N3 * 32;
    const long long y3n = (long long)BATCH * N4 * 64;
    const long long y4n = (long long)BATCH * N5 * 128;

    dim3 thr(256);

    // ---------- layer 0: scale + scatter ----------
    zero_f32<<<4096, thr, 0, stream>>>(bufA, y0n);
    layer0_scatter<<<8192, thr, 0, stream>>>(x, Wl[0], par[0], bufA,
                                             (long long)BATCH * N0);
    bias_lrelu<8><<<4096, thr, 0, stream>>>(bufA, bl[0], y0n);

    // ---------- layer 1: (64,32768,8) -> (64,8192,16) ----------
    zero_f32<<<4096, thr, 0, stream>>>(bufB, y1n);
    fgl_wmma<8, 16><<<N1, 128, 0, stream>>>(bufA, Wl[1], par[1], bufB, N1, N2);
    bias_lrelu<16><<<4096, thr, 0, stream>>>(bufB, bl[1], y1n);

    // ---------- layer 2: (64,8192,16) -> (64,2048,32) ----------
    zero_f32<<<2048, thr, 0, stream>>>(bufA, y2n);
    fgl_wmma<16, 32><<<N2, 128, 0, stream>>>(bufB, Wl[2], par[2], bufA, N2, N3);
    bias_lrelu<32><<<2048, thr, 0, stream>>>(bufA, bl[2], y2n);

    // ---------- layer 3: (64,2048,32) -> (64,512,64) ----------
    zero_f32<<<1024, thr, 0, stream>>>(bufB, y3n);
    fgl_wmma<32, 64><<<N3, 128, 0, stream>>>(bufA, Wl[3], par[3], bufB, N3, N4);
    bias_lrelu<64><<<1024, thr, 0, stream>>>(bufB, bl[3], y3n);

    // ---------- layer 4: (64,512,64) -> (64,128,128) ----------
    zero_f32<<<512, thr, 0, stream>>>(bufA, y4n);
    fgl_wmma<64, 128><<<N4, 128, 0, stream>>>(bufB, Wl[4], par[4], bufA, N4, N5);
    bias_lrelu<128><<<512, thr, 0, stream>>>(bufA, bl[4], y4n);

    // ---------- head ----------
    fc_head<<<BATCH, 256, 0, stream>>>(bufA, fc_w, fc_b, out);

    (void)in_sizes; (void)n_in; (void)out_size; (void)ws_size;
}